// MomentConv2d_71992241815664
// MI455X (gfx1250) — compile-verified
//
#include <hip/hip_runtime.h>

// MomentConv2d on MI455X (gfx1250, wave32).
//
// Shapes: B=32, CIN=16, COUT=32, H=W=14, K=5, S=2 -> OH=OW=5, P=25.
// cin-sum hoisted before both convs (kernels are cin-independent) => problem
// is HBM-bound on one pass over C (78.7 MB ~ 3.4 us at 23.3 TB/s). Step-1
// conv AND the mean path run as dense GEMMs W(32x25) x P(25x25) per "site"
// using v_wmma_f32_16x16x4_f32; source rows are staged global->LDS with
// GLOBAL_LOAD_ASYNC_TO_LDS_B32 (ASYNCcnt); intermediate T (20 MB) stays in
// the 192 MB L2. B-fragment gather is branch-free (cndmask to a zeroed LDS
// pad slot) so the WMMA loop is straight-line.

typedef float v2f __attribute__((ext_vector_type(2)));
typedef float v8f __attribute__((ext_vector_type(8)));

#define B_    32
#define CIN_  16
#define COUT_ 32
#define H_    14
#define W_    14
#define KW_   5
#define P_    25      // OH*OW
#define HW_   196     // H*W
#define HW2_  38416   // HW*HW
#define KK_   25      // K*K
#define ROWSTRIDE_ 200  // 196 data + 4 zero-pad floats per wave row in LDS
#define NSITE_COV_  (B_ * HW_)        // 6272
#define NSITE_ALL_  (NSITE_COV_ + B_) // 6304 = 788 * 8

// ---------------- Kernel 1: Cs[b,xy] = sum_cin C[b,cin,xy] ------------------
// Pure streaming reduction: 78.7 MB read, 4.9 MB write; coalesced lanes,
// bandwidth bound — this is the kernel that sets the overall runtime floor.
__global__ void __launch_bounds__(256) reduce_cin_kernel(
    const float* __restrict__ C, float* __restrict__ Cs) {
  int j = blockIdx.x * 256 + threadIdx.x;
  if (j >= B_ * HW2_) return;
  int b  = j / HW2_;
  int xy = j - b * HW2_;
  const float* p = C + (size_t)b * CIN_ * HW2_ + xy;
  float s = 0.f;
#pragma unroll
  for (int c = 0; c < CIN_; ++c) s += p[(size_t)c * HW2_];
  Cs[j] = s;
}

// ---------------- Kernel 1b: Us[b,x] = sum_cin u[b,cin,x] -------------------
__global__ void __launch_bounds__(256) reduce_u_kernel(
    const float* __restrict__ u, float* __restrict__ Us) {
  int j = blockIdx.x * 256 + threadIdx.x;
  if (j >= B_ * HW_) return;
  int b = j / HW_;
  int x = j - b * HW_;
  const float* p = u + (size_t)b * CIN_ * HW_ + x;
  float s = 0.f;
#pragma unroll
  for (int c = 0; c < CIN_; ++c) s += p[(size_t)c * HW_];
  Us[j] = s;
}

// ---------------- Kernel 2: conv-as-GEMM via WMMA ---------------------------
// One wave per site. Sites 0..6271: s=(b,x), row=Cs[b,x,:],
//   T[s,i,q] = sum_k w[i,k] * row[(2qh+kh)*14 + 2qw+kw]   (step-1 cov conv)
// Sites 6272..6303: b=s-6272, row=Us[b,:], output -> out_mean[b,i,p]
//   (mean path: identical GEMM, conv of the cin-summed image).
// GEMM: M=32(i, 2 tiles), N=25(q, 2 tiles padded), K=25 (7 steps of 4).
// f32 16x16x4 fragment layout: lane half selects K pair, vector component
// selects within pair; M/N = lane&15.
__global__ void __launch_bounds__(256) step1_wmma_kernel(
    const float* __restrict__ Cs, const float* __restrict__ Us,
    const float* __restrict__ w, float* __restrict__ T,
    float* __restrict__ out_mean) {
  __shared__ float sh[8 * ROWSTRIDE_];
  const int lane = threadIdx.x & 31;
  const int wave = threadIdx.x >> 5;
  const int site = blockIdx.x * 8 + wave;  // 0..6303, exact cover (788 blocks)

  const bool is_mean = site >= NSITE_COV_;
  const float* row = is_mean ? (Us + (size_t)(site - NSITE_COV_) * HW_)
                             : (Cs + (size_t)site * HW_);
  float* shrow = sh + wave * ROWSTRIDE_;

  // Stage the 196-float source row global -> LDS with the gfx1250 async path
  // (no VGPR round-trip, tracked by ASYNCcnt).
  for (int t = lane; t < HW_; t += 32) {
    unsigned lds_addr = (unsigned)(uintptr_t)(shrow + t);  // wave-relative LDS byte addr
    unsigned long long gaddr = (unsigned long long)(uintptr_t)(row + t);
    asm volatile("global_load_async_to_lds_b32 %0, %1, off"
                 :: "v"(lds_addr), "v"(gaddr) : "memory");
  }
  if (lane < 4) shrow[HW_ + lane] = 0.f;  // zero pad slot for invalid gathers
  asm volatile("s_wait_asynccnt 0" ::: "memory");
  __syncthreads();

  const int hsel = lane >> 4;  // 0: K in {4k,4k+1}, 1: K in {4k+2,4k+3}
  const int n    = lane & 15;  // M (A) / N (B,D) index within tile

  // Per-lane output-position bases (q0 always valid; q1 valid iff n < 9).
  const int q0 = n;
  const int q1 = 16 + n;
  const bool q1v = (q1 < P_);
  const int q1c = q1v ? q1 : 0;
  const int base0 = (q0 / 5) * (2 * W_) + (q0 % 5) * 2;
  const int base1 = (q1c / 5) * (2 * W_) + (q1c % 5) * 2;

  // A fragments (weights), branch-free: clamp index, mask value.
  v2f af[2][7];
#pragma unroll
  for (int it = 0; it < 2; ++it) {
    int i = it * 16 + n;
#pragma unroll
    for (int ks = 0; ks < 7; ++ks) {
      int k0 = ks * 4 + hsel * 2;
      int k1 = k0 + 1;
      int k0c = (k0 < KK_) ? k0 : 0;
      int k1c = (k1 < KK_) ? k1 : 0;
      float a0 = w[i * KK_ + k0c];
      float a1 = w[i * KK_ + k1c];
      v2f a;
      a.x = (k0 < KK_) ? a0 : 0.f;
      a.y = (k1 < KK_) ? a1 : 0.f;
      af[it][ks] = a;
    }
  }

  v8f acc00 = {}, acc01 = {}, acc10 = {}, acc11 = {};

#pragma unroll
  for (int ks = 0; ks < 7; ++ks) {
    int k0 = ks * 4 + hsel * 2;
    int k1 = k0 + 1;
    int k0c = (k0 < KK_) ? k0 : 0;
    int k1c = (k1 < KK_) ? k1 : 0;
    int koff0 = (k0c / 5) * W_ + (k0c % 5);
    int koff1 = (k1c / 5) * W_ + (k1c % 5);
    bool kv0 = (k0 < KK_), kv1 = (k1 < KK_);
    // Unconditional ds_loads; invalid lanes read the zeroed pad slot.
    v2f b0, b1;
    b0.x = shrow[kv0 ? (base0 + koff0) : HW_];
    b0.y = shrow[kv1 ? (base0 + koff1) : HW_];
    b1.x = shrow[(kv0 && q1v) ? (base1 + koff0) : HW_];
    b1.y = shrow[(kv1 && q1v) ? (base1 + koff1) : HW_];
    // 8-arg form: (neg_a, A, neg_b, B, c_mod, C, reuse_a, reuse_b)
    acc00 = __builtin_amdgcn_wmma_f32_16x16x4_f32(false, af[0][ks], false, b0, (short)0, acc00, false, false);
    acc10 = __builtin_amdgcn_wmma_f32_16x16x4_f32(false, af[1][ks], false, b0, (short)0, acc10, false, false);
    acc01 = __builtin_amdgcn_wmma_f32_16x16x4_f32(false, af[0][ks], false, b1, (short)0, acc01, false, false);
    acc11 = __builtin_amdgcn_wmma_f32_16x16x4_f32(false, af[1][ks], false, b1, (short)0, acc11, false, false);
  }

  // D layout: lanes 0-15 -> M=vgpr(0..7), lanes 16-31 -> M=8+vgpr; N=lane&15.
  // Cov sites -> T[site*800 + i*25 + q]; mean sites -> out_mean[b*800 + i*25 + p].
  float* outp = is_mean ? (out_mean + (size_t)(site - NSITE_COV_) * (COUT_ * P_))
                        : (T + (size_t)site * (COUT_ * P_));
#pragma unroll
  for (int it = 0; it < 2; ++it) {
    // qt = 0: q = n < 16, always valid -> unconditional stores.
    v8f a = (it == 0) ? acc00 : acc10;
#pragma unroll
    for (int r = 0; r < 8; ++r) {
      int i = it * 16 + hsel * 8 + r;
      outp[i * P_ + q0] = a[r];
    }
    // qt = 1: q = 16 + n, valid iff n < 9.
    v8f c = (it == 0) ? acc01 : acc11;
    if (q1v) {
#pragma unroll
      for (int r = 0; r < 8; ++r) {
        int i = it * 16 + hsel * 8 + r;
        outp[i * P_ + q1] = c[r];
      }
    }
  }
}

// ---------------- Kernel 3: step-2 depthwise conv -> out_cov ----------------
// out_cov[b,i,p,q] = sum_k1 w[i,k1] * T[b, x=2p+k1, i, q]. 640K outputs,
// 25 VALU FMAs each; T streamed from L2 (20 MB resident).
__global__ void __launch_bounds__(256) step2_kernel(
    const float* __restrict__ T, const float* __restrict__ w,
    float* __restrict__ out_cov) {
  int idx = blockIdx.x * 256 + threadIdx.x;
  if (idx >= B_ * COUT_ * P_ * P_) return;
  int q = idx % P_;
  int p = (idx / P_) % P_;
  int i = (idx / (P_ * P_)) % COUT_;
  int b = idx / (COUT_ * P_ * P_);
  int ph = p / 5, pw = p - ph * 5;
  float s = 0.f;
#pragma unroll
  for (int kh = 0; kh < KW_; ++kh) {
#pragma unroll
    for (int kw = 0; kw < KW_; ++kw) {
      int x = (2 * ph + kh) * W_ + (2 * pw + kw);
      s += w[i * KK_ + kh * KW_ + kw] *
           T[((size_t)(b * HW_ + x) * COUT_ + i) * P_ + q];
    }
  }
  out_cov[idx] = s;
}

extern "C" void kernel_launch(void* const* d_in, const int* in_sizes, int n_in,
                              void* d_out, int out_size, void* d_ws, size_t ws_size,
                              hipStream_t stream) {
  const float* u = (const float*)d_in[0];  // [32,16,14,14]
  const float* C = (const float*)d_in[1];  // [32,16,14,14,14,14]
  const float* w = (const float*)d_in[2];  // [32,1,5,5]

  float* out_mean = (float*)d_out;                    // 25600 floats
  float* out_cov  = (float*)d_out + B_ * COUT_ * P_;  // 640000 floats

  // Workspace: Cs (4.92 MB) + T (20.07 MB) + Us (25 KB) ~ 25 MB.
  float* Cs = (float*)d_ws;
  float* T  = Cs + (size_t)B_ * HW2_;
  float* Us = T + (size_t)NSITE_COV_ * (COUT_ * P_);

  // 1) cin reductions (the HBM-bound pass over C, plus tiny u reduction).
  reduce_cin_kernel<<<(B_ * HW2_) / 256, 256, 0, stream>>>(C, Cs);
  reduce_u_kernel<<<(B_ * HW_ + 255) / 256, 256, 0, stream>>>(u, Us);

  // 2) WMMA GEMMs: 6272 cov sites + 32 mean sites, 8 waves/block -> 788 blocks.
  step1_wmma_kernel<<<NSITE_ALL_ / 8, 256, 0, stream>>>(Cs, Us, w, T, out_mean);

  // 3) step-2 conv -> out_cov (640,000 elements).
  step2_kernel<<<(B_ * COUT_ * P_ * P_ + 255) / 256, 256, 0, stream>>>(T, w, out_cov);
}